// ZBLRepulsionEnergy_24945170055212
// MI455X (gfx1250) — compile-verified
//
#include <hip/hip_runtime.h>
#include <hip/hip_bf16.h>

#define KEHALF_F  7.199822675975274f
#define CUTOFF_F  10.0f

// ---------------- CDNA5 async global->LDS staging (guarded) ----------------
#if defined(__HIP_DEVICE_COMPILE__) && __has_builtin(__builtin_amdgcn_global_load_async_to_lds_b32)
#define HAVE_ASYNC_LDS 1
#else
#define HAVE_ASYNC_LDS 0
#endif

#if defined(__HIP_DEVICE_COMPILE__) && __has_builtin(__builtin_amdgcn_global_load_async_to_lds_b128)
#define HAVE_ASYNC_B128 1
#else
#define HAVE_ASYNC_B128 0
#endif

#define GLOBAL_AS __attribute__((address_space(1)))
#define LDS_AS    __attribute__((address_space(3)))

typedef int v4i __attribute__((__vector_size__(4 * sizeof(int))));

__device__ __forceinline__ void async_cp_b32(const void* g, void* l) {
#if HAVE_ASYNC_LDS
    __builtin_amdgcn_global_load_async_to_lds_b32(
        (GLOBAL_AS int*)(void*)g, (LDS_AS int*)l, 0, 0);
#else
    *(volatile int*)l = *(const int*)g;   // wave-synchronous fallback
#endif
}

#if HAVE_ASYNC_B128
__device__ __forceinline__ void async_cp_b128(const void* g, void* l) {
    __builtin_amdgcn_global_load_async_to_lds_b128(
        (GLOBAL_AS v4i*)(void*)g, (LDS_AS v4i*)l, 0, 0);
}
#endif

#if HAVE_ASYNC_LDS
#  if __has_builtin(__builtin_amdgcn_s_wait_asynccnt)
#    define ASYNC_WAIT(n) do { __builtin_amdgcn_s_wait_asynccnt(n); \
                               asm volatile("" ::: "memory"); } while (0)
#  else
#    define ASYNC_WAIT(n) asm volatile("s_wait_asynccnt " #n ::: "memory")
#  endif
#else
#  define ASYNC_WAIT(n) asm volatile("" ::: "memory")
#endif

// Number of async ops issued per staged tile (wait threshold for pipelining).
#if HAVE_ASYNC_B128
#define STAGE_OPS 3
#else
#define STAGE_OPS 5
#endif

// ---------------- math helpers ----------------
__device__ __forceinline__ float softplus_f(float x) {
    return (x > 20.0f) ? x : log1pf(__expf(x));
}

struct EdgeParams {
    float spd;                       // softplus(adiv)
    float sa0, sa1, sa2, sa3;        // softplus(a_vector[k])
    float cn0, cn1, cn2, cn3;        // L1-normalized softplus(c_vector[k])
};

__device__ __forceinline__ EdgeParams load_params(const float* adiv,
                                                  const float* a_vec,
                                                  const float* c_vec) {
    EdgeParams p;
    p.spd = softplus_f(adiv[0]);
    p.sa0 = softplus_f(a_vec[0]); p.sa1 = softplus_f(a_vec[1]);
    p.sa2 = softplus_f(a_vec[2]); p.sa3 = softplus_f(a_vec[3]);
    float c0 = softplus_f(c_vec[0]), c1 = softplus_f(c_vec[1]);
    float c2 = softplus_f(c_vec[2]), c3 = softplus_f(c_vec[3]);
    float inv = __frcp_rn(fabsf(c0) + fabsf(c1) + fabsf(c2) + fabsf(c3));
    p.cn0 = c0 * inv; p.cn1 = c1 * inv; p.cn2 = c2 * inv; p.cn3 = c3 * inv;
    return p;
}

__device__ __forceinline__ float edge_energy(float rx, float ry, float rz,
                                             float2 zi, float2 zj,
                                             const EdgeParams& p) {
    float d2 = fmaf(rx, rx, fmaf(ry, ry, rz * rz));
    float d  = sqrtf(d2);
    float a  = (zi.y + zj.y) * p.spd;
    float ad = a * d;
    float s  = p.cn0 * __expf(-p.sa0 * ad) + p.cn1 * __expf(-p.sa1 * ad)
             + p.cn2 * __expf(-p.sa2 * ad) + p.cn3 * __expf(-p.sa3 * ad);
    float u  = d * (1.0f / CUTOFF_F);
    float u3 = u * u * u;
    float fc = 1.0f - u3 * fmaf(u, fmaf(6.0f, u, -15.0f), 10.0f); // 1-10u^3+15u^4-6u^5
    fc = (d < CUTOFF_F) ? fc : 0.0f;
    return __fdividef(KEHALF_F * s * fc * zi.x * zj.x, d);
}

// ---------------- kernels ----------------
__global__ void __launch_bounds__(256) zbl_zero_kernel(float* y_atom, int N,
                                                       float* y, int M) {
    int t = blockIdx.x * blockDim.x + threadIdx.x;
    if (t < N) y_atom[t] = 0.0f;
    if (t < M) y[t] = 0.0f;
}

__global__ void __launch_bounds__(256) zbl_pz_kernel(const float* __restrict__ Z,
                                                     const float* __restrict__ apow,
                                                     float2* __restrict__ pz, int N) {
    float p = softplus_f(apow[0]);
    int n = blockIdx.x * blockDim.x + threadIdx.x;
    if (n < N) {
        float z = Z[n];
        pz[n] = make_float2(z, __powf(z, p));
    }
}

#define WAVES_PER_BLOCK 8

__global__ void __launch_bounds__(256) zbl_edge_kernel(
    const float2* __restrict__ pz,
    const float*  __restrict__ r_ij,
    const int*    __restrict__ idx_i,
    const int*    __restrict__ idx_j,
    const float*  __restrict__ adiv,
    const float*  __restrict__ a_vec,
    const float*  __restrict__ c_vec,
    float*        __restrict__ y_atom,
    int E)
{
    // Per-wave double-buffered LDS tiles: 32 edges per tile.
    // Slices per wave: r = 384B, idx = 128B -> all 16B-aligned for b128.
    __shared__ __align__(16) float s_r [2][WAVES_PER_BLOCK][96];
    __shared__ __align__(16) int   s_ii[2][WAVES_PER_BLOCK][32];
    __shared__ __align__(16) int   s_jj[2][WAVES_PER_BLOCK][32];

    const EdgeParams prm = load_params(adiv, a_vec, c_vec);

    const int wave = threadIdx.x >> 5;    // wave32
    const int lane = threadIdx.x & 31;
    const int nslots = gridDim.x * WAVES_PER_BLOCK;
    const int slot   = blockIdx.x * WAVES_PER_BLOCK + wave;
    const int ntiles = E >> 5;            // full 32-edge tiles

    auto stage = [&](int tile, int b) {
        const int base = tile << 5;
        const float* rg = r_ij + (size_t)base * 3;
        float* rb = &s_r [b][wave][0];
        int*   ib = &s_ii[b][wave][0];
        int*   jb = &s_jj[b][wave][0];
#if HAVE_ASYNC_B128
        if (lane < 24) async_cp_b128(rg + lane * 4, rb + lane * 4);
        if (lane < 8)  async_cp_b128(idx_i + base + lane * 4, ib + lane * 4);
        if (lane < 8)  async_cp_b128(idx_j + base + lane * 4, jb + lane * 4);
#else
        async_cp_b32(rg + lane,        rb + lane);
        async_cp_b32(rg + 32 + lane,   rb + 32 + lane);
        async_cp_b32(rg + 64 + lane,   rb + 64 + lane);
        async_cp_b32(idx_i + base + lane, ib + lane);
        async_cp_b32(idx_j + base + lane, jb + lane);
#endif
    };

    int buf = 0;
    int t = slot;
    if (t < ntiles) stage(t, 0);
    for (; t < ntiles; t += nslots) {
        const int tn = t + nslots;
        if (tn < ntiles) { stage(tn, buf ^ 1); ASYNC_WAIT(STAGE_OPS); }
        else             { ASYNC_WAIT(0); }

        const float* rb = &s_r [buf][wave][0];
        const int*   ib = &s_ii[buf][wave][0];
        const int*   jb = &s_jj[buf][wave][0];
        float rx = rb[lane * 3 + 0];
        float ry = rb[lane * 3 + 1];
        float rz = rb[lane * 3 + 2];
        int i = ib[lane];
        int j = jb[lane];
        float2 zi = pz[i];
        float2 zj = pz[j];
        float e = edge_energy(rx, ry, rz, zi, zj, prm);
        atomicAdd(&y_atom[i], e);
        buf ^= 1;
    }

    // Tail edges (E % 32) handled directly from global memory.
    for (int e = (ntiles << 5) + blockIdx.x * blockDim.x + threadIdx.x;
         e < E; e += gridDim.x * blockDim.x) {
        float rx = r_ij[(size_t)e * 3 + 0];
        float ry = r_ij[(size_t)e * 3 + 1];
        float rz = r_ij[(size_t)e * 3 + 2];
        int i = idx_i[e];
        int j = idx_j[e];
        float en = edge_energy(rx, ry, rz, pz[i], pz[j], prm);
        atomicAdd(&y_atom[i], en);
    }
}

__global__ void __launch_bounds__(256) zbl_mol_kernel(const float* __restrict__ y_atom,
                                                      const int*   __restrict__ idx_m,
                                                      float*       __restrict__ y,
                                                      int N) {
    int n = blockIdx.x * blockDim.x + threadIdx.x;
    if (n < N) atomicAdd(&y[idx_m[n]], y_atom[n]);
}

// ---------------- launch ----------------
extern "C" void kernel_launch(void* const* d_in, const int* in_sizes, int n_in,
                              void* d_out, int out_size, void* d_ws, size_t ws_size,
                              hipStream_t stream) {
    const float* Z     = (const float*)d_in[0];
    const float* r_ij  = (const float*)d_in[1];
    const int*   idx_i = (const int*)d_in[2];
    const int*   idx_j = (const int*)d_in[3];
    const int*   idx_m = (const int*)d_in[4];
    const float* adiv  = (const float*)d_in[5];
    const float* apow  = (const float*)d_in[6];
    const float* a_vec = (const float*)d_in[7];
    const float* c_vec = (const float*)d_in[8];
    float* y = (float*)d_out;

    const int N = in_sizes[0];
    const int E = in_sizes[2];

    float* y_atom = (float*)d_ws;
    size_t off = ((size_t)N * sizeof(float) + 15) & ~(size_t)15;
    float2* pz = (float2*)((char*)d_ws + off);

    const int tpb = 256;
    const int gN = (N + tpb - 1) / tpb;

    zbl_zero_kernel<<<gN, tpb, 0, stream>>>(y_atom, N, y, out_size);
    zbl_pz_kernel<<<gN, tpb, 0, stream>>>(Z, apow, pz, N);
    zbl_edge_kernel<<<1024, 256, 0, stream>>>(pz, r_ij, idx_i, idx_j,
                                              adiv, a_vec, c_vec, y_atom, E);
    zbl_mol_kernel<<<gN, tpb, 0, stream>>>(y_atom, idx_m, y, N);
}